// ThreatDetectorGNN_58961311040081
// MI455X (gfx1250) — compile-verified
//
#include <hip/hip_runtime.h>

// ---- WMMA vector types (wave32, gfx1250) ----
typedef __attribute__((ext_vector_type(2))) float v2f;   // A/B frag of V_WMMA_F32_16X16X4_F32
typedef __attribute__((ext_vector_type(8))) float v8f;   // 16x16 f32 C/D accumulator

// =====================================================================
// Degree kernels: deg[i] = 1 (self loop) + #edges with dst==i; then rsqrt
// =====================================================================
__global__ void deg_init_kernel(float* __restrict__ deg, int N) {
    int i = blockIdx.x * blockDim.x + threadIdx.x;
    if (i < N) deg[i] = 1.0f;   // self-loop contribution
}

__global__ void deg_count_kernel(const int* __restrict__ dst, float* __restrict__ deg, int E) {
    int e = blockIdx.x * blockDim.x + threadIdx.x;
    if (e < E) atomicAdd(&deg[dst[e]], 1.0f);
}

__global__ void rsqrt_inplace_kernel(float* __restrict__ deg, int N) {
    int i = blockIdx.x * blockDim.x + threadIdx.x;
    if (i < N) deg[i] = rsqrtf(deg[i]);
}

// =====================================================================
// FP32 WMMA GEMM: out[M,FOUT] = (RELU ? relu(A) : A)[M,K] @ W[K,FOUT]
// One block per 16-row stripe; wave w (= threadIdx.x>>5) owns col tile w.
// blockDim.x = 32 * (FOUT/16).  Requires M%16==0 (M=100000 -> 6250 tiles),
// K%4==0, FOUT%16==0.  K-loop fully unrolled -> 32 static v_wmma.
//
// A 16x4 f32 frag : lane l -> row m = l&15 ; VGPRs = K = kb, kb+1,
//                   kb = (l>>4)*2 (lanes 0-15: K0/K1, lanes 16-31: K2/K3)
// B 4x16  f32 frag: lane l -> col n = l&15 ; VGPRs = K = kb, kb+1
// C/D 16x16 f32   : VGPR v -> row (v + 8*(l>>4)), col n = l&15
// =====================================================================
template <int K, int FOUT, bool RELU>
__global__ __launch_bounds__(32 * (FOUT / 16))
void gcn_gemm_wmma_f32(const float* __restrict__ A,
                       const float* __restrict__ W,
                       float* __restrict__ out) {
    const int lane = threadIdx.x & 31;
    const int mn   = lane & 15;             // row of A-frag / col of B-frag
    const int sel  = lane >> 4;             // which K-pair this half-wave holds
    const size_t row0 = (size_t)blockIdx.x * 16;
    const int col0 = (threadIdx.x >> 5) * 16;   // wave id -> col tile

    v8f c = {0.f, 0.f, 0.f, 0.f, 0.f, 0.f, 0.f, 0.f};

    // A pairs: pair index (k0/2 + sel) = 2*kk + sel  (8B aligned)
    const v2f* __restrict__ apair =
        reinterpret_cast<const v2f*>(A + (row0 + mn) * K) + sel;
    // B: rows kb, kb+1 of W, column col0+mn
    const float* __restrict__ bp = W + (size_t)(sel * 2) * FOUT + col0 + mn;

#pragma unroll
    for (int kk = 0; kk < K / 4; ++kk) {
        v2f a = apair[2 * kk];
        if (RELU) {
            a.x = fmaxf(a.x, 0.0f);
            a.y = fmaxf(a.y, 0.0f);
        }
        v2f b;
        b.x = bp[(size_t)(4 * kk)     * FOUT];
        b.y = bp[(size_t)(4 * kk + 1) * FOUT];
        // D = A*B + C  (fp32 matrix core, 16x16x4)
        c = __builtin_amdgcn_wmma_f32_16x16x4_f32(
                /*neg_a=*/false, a, /*neg_b=*/false, b,
                /*c_mod=*/(short)0, c, /*reuse_a=*/false, /*reuse_b=*/false);
    }

    float* __restrict__ obase = out + row0 * FOUT + col0 + mn;
#pragma unroll
    for (int v = 0; v < 8; ++v) {
        obase[(size_t)(v + 8 * sel) * FOUT] = c[v];
    }
}

// =====================================================================
// agg[n,f] = bias[f] + h[n,f] * dis[n]^2      (self-loop term + bias)
// =====================================================================
template <int LOGF>
__global__ void agg_self_init_kernel(const float* __restrict__ h,
                                     const float* __restrict__ dis,
                                     const float* __restrict__ bias,
                                     float* __restrict__ agg,
                                     size_t total) {
    size_t i = (size_t)blockIdx.x * blockDim.x + threadIdx.x;
    if (i >= total) return;
    int n = (int)(i >> LOGF);
    int f = (int)(i & ((1u << LOGF) - 1u));
    float d = dis[n];
    agg[i] = bias[f] + h[i] * d * d;
}

// =====================================================================
// Edge scatter: one wave per edge.
//   agg[dst] += h[src] * dis[src]*dis[dst]   (per-lane fp32 global atomics)
// F=128 -> float4 per lane; F=64 -> float2 per lane (coalesced gathers).
// =====================================================================
template <int F>
__global__ void agg_edges_kernel(const float* __restrict__ h,
                                 const int* __restrict__ src,
                                 const int* __restrict__ dst,
                                 const float* __restrict__ dis,
                                 float* __restrict__ agg,
                                 int E) {
    const int wave = (int)(((size_t)blockIdx.x * blockDim.x + threadIdx.x) >> 5);
    const int lane = threadIdx.x & 31;
    if (wave >= E) return;

    const int   s    = src[wave];
    const int   d    = dst[wave];
    const float norm = dis[s] * dis[d];

    const float* __restrict__ hs = h   + (size_t)s * F;
    float* __restrict__       ad = agg + (size_t)d * F;

    if constexpr (F == 128) {
        float4 v = reinterpret_cast<const float4*>(hs)[lane];
        atomicAdd(&ad[lane * 4 + 0], v.x * norm);
        atomicAdd(&ad[lane * 4 + 1], v.y * norm);
        atomicAdd(&ad[lane * 4 + 2], v.z * norm);
        atomicAdd(&ad[lane * 4 + 3], v.w * norm);
    } else {  // F == 64
        float2 v = reinterpret_cast<const float2*>(hs)[lane];
        atomicAdd(&ad[lane * 2 + 0], v.x * norm);
        atomicAdd(&ad[lane * 2 + 1], v.y * norm);
    }
}

// =====================================================================
// Host-side launcher
// =====================================================================
extern "C" void kernel_launch(void* const* d_in, const int* in_sizes, int n_in,
                              void* d_out, int out_size, void* d_ws, size_t ws_size,
                              hipStream_t stream) {
    const float* x     = (const float*)d_in[0];     // [N,128]
    const int*   edges = (const int*)  d_in[1];     // [2,E] row-major
    const float* W1    = (const float*)d_in[2];     // [128,128]
    const float* b1    = (const float*)d_in[3];     // [128]
    const float* W2    = (const float*)d_in[4];     // [128,64]
    const float* b2    = (const float*)d_in[5];     // [64]
    float*       out   = (float*)d_out;             // [N,64]

    const int N = in_sizes[0] / 128;                // 100000 (divisible by 16)
    const int E = in_sizes[1] / 2;                  // 1600000

    const int* src = edges;
    const int* dst = edges + E;

    // Workspace layout (bytes): dis @0 (N*4), bufA @4MB (N*128*4),
    // bufB @4MB+N*128*4.  Total ~107 MB.
    char*  ws   = (char*)d_ws;
    float* dis  = (float*)(ws);
    float* bufA = (float*)(ws + (size_t)4 * 1024 * 1024);                 // h1, then h2
    float* bufB = (float*)(ws + (size_t)4 * 1024 * 1024
                              + (size_t)N * 128 * sizeof(float));         // agg1

    const int T = 256;

    // --- degrees -> deg_inv_sqrt (in place in `dis`) ---
    deg_init_kernel<<<(N + T - 1) / T, T, 0, stream>>>(dis, N);
    deg_count_kernel<<<(E + T - 1) / T, T, 0, stream>>>(dst, dis, E);
    rsqrt_inplace_kernel<<<(N + T - 1) / T, T, 0, stream>>>(dis, N);

    // --- layer 1: h1 = x @ W1 (fp32 WMMA, 8 waves/block = 8 col tiles) ---
    gcn_gemm_wmma_f32<128, 128, false><<<N / 16, 32 * 8, 0, stream>>>(x, W1, bufA);

    // agg1 = b1 + h1*dis^2, then scatter edges
    {
        size_t total = (size_t)N * 128;
        agg_self_init_kernel<7><<<(unsigned)((total + T - 1) / T), T, 0, stream>>>(
            bufA, dis, b1, bufB, total);
        agg_edges_kernel<128><<<(unsigned)(((size_t)E * 32 + T - 1) / T), T, 0, stream>>>(
            bufA, src, dst, dis, bufB, E);
    }

    // --- layer 2: h2 = relu(agg1) @ W2 (relu fused into A-load, 4 waves/block) ---
    gcn_gemm_wmma_f32<128, 64, true><<<N / 16, 32 * 4, 0, stream>>>(bufB, W2, bufA);

    // out = b2 + h2*dis^2, then scatter edges
    {
        size_t total = (size_t)N * 64;
        agg_self_init_kernel<6><<<(unsigned)((total + T - 1) / T), T, 0, stream>>>(
            bufA, dis, b2, out, total);
        agg_edges_kernel<64><<<(unsigned)(((size_t)E * 32 + T - 1) / T), T, 0, stream>>>(
            bufA, src, dst, dis, out, E);
    }
}